// OrthoLSTMCell_31061203485011
// MI455X (gfx1250) — compile-verified
//
#include <hip/hip_runtime.h>
#include <stdint.h>

// ---------------------------------------------------------------------------
// OrthoLSTMCell for MI455X (gfx1250, wave32, WMMA + async global->LDS DMA)
//   B = 16384 rows, H = 1024 hidden.
//   Z_n = d@W_n^T + b_n + h@U_n^T  (5 gates) => 343.6 GFLOP fp32, compute-bound.
//   - V_WMMA_F32_16X16X4_F32 for the fp32 matrix path (matches reference dtype)
//   - GLOBAL_LOAD_ASYNC_TO_LDS_B128 (ASYNCcnt) + double-buffered LDS so the
//     next K-tile streams into LDS while WMMAs consume the current one.
// ---------------------------------------------------------------------------

#define Bdim 16384
#define Hdim 1024
#define BM   64     // block tile M
#define BN   32     // block tile N
#define KT   32     // K tile staged in LDS per step
#define LDA  36     // padded LDS row stride (dwords); 36*4=144 B = 16B-aligned rows
#define LDB  36

typedef __attribute__((ext_vector_type(2))) float v2f;   // 16x4 fp32 A/B frag: 2 VGPRs
typedef __attribute__((ext_vector_type(8))) float v8f;   // 16x16 fp32 C/D frag: 8 VGPRs

struct GatePtrs {
  const float* W[5];   // [H,H], row n contiguous in k
  const float* U[5];   // [H,H]
  const float* b[5];   // [H]
};

__device__ __forceinline__ float fast_sigmoid(float x) {
  return 1.0f / (1.0f + __expf(-x));
}
__device__ __forceinline__ float fast_tanh(float x) {
  return 2.0f / (1.0f + __expf(-2.0f * x)) - 1.0f;
}

// Low 32 bits of a generic pointer to __shared__ = LDS byte address.
__device__ __forceinline__ uint32_t lds_off(const void* p) {
  return (uint32_t)(uintptr_t)p;
}

// Async DMA: 16 bytes per lane, global -> LDS, no VGPR round-trip (ASYNCcnt).
__device__ __forceinline__ void async_copy16(uint32_t lds_byte, const float* gaddr) {
  asm volatile("global_load_async_to_lds_b128 %0, %1, off"
               :: "v"(lds_byte), "v"(gaddr)
               : "memory");
}
__device__ __forceinline__ void wait_async_le7() {
  asm volatile("s_wait_asynccnt 0x7" ::: "memory");
}
__device__ __forceinline__ void wait_async_le0() {
  asm volatile("s_wait_asynccnt 0x0" ::: "memory");
}

// ---------------------------------------------------------------------------
// Kernel 1: all-gate GEMM + fused gate activations.
//   256 threads (8 waves) per block; 64x32 tile of all 5 gates.
//   Wave w: wm = w>>1 (M sub-tile), wn = w&1 (N sub-tile); one 16x16 fragment
//   per gate, bias-initialized. Double-buffered async staging.
// ---------------------------------------------------------------------------
__global__ __launch_bounds__(256)
void ortho_lstm_gates_gemm(const float* __restrict__ dt,
                           const float* __restrict__ hprev,
                           const float* __restrict__ cprev,
                           GatePtrs gp,
                           float* __restrict__ out_c,
                           float* __restrict__ ws_g,
                           float* __restrict__ ws_o)
{
  __shared__ float As[2][BM * LDA];        // 2 x  9.2 KB
  __shared__ float Bs[2][5 * BN * LDB];    // 2 x 23.0 KB

  const int tid  = threadIdx.x;
  const int wave = tid >> 5;
  const int lane = tid & 31;
  const int lh   = lane >> 4;   // half-wave: K pair / M+8 per ISA frag layout
  const int l16  = lane & 15;
  const int wm   = wave >> 1;   // 0..3
  const int wn   = wave & 1;    // 0..1

  const int m0 = blockIdx.y * BM;
  const int n0 = blockIdx.x * BN;
  const int nG = n0 + wn * 16 + l16;

  // Bias-initialized accumulators (bias depends only on N, constant over v).
  v8f acc[5];
#pragma unroll
  for (int g = 0; g < 5; ++g) {
    const float bb = gp.b[g][nG];
    acc[g] = (v8f){bb, bb, bb, bb, bb, bb, bb, bb};
  }

  const int NT = 2 * (Hdim / KT);   // 64 K-tiles total (phase 0: d/W, 1: h/U)

  // Issue one K-tile's staging as 7 async b128 ops per wave (2 for A, 5 for B).
  // Gate index is the unroll index -> compile-time indexing of gp arrays.
  auto issue = [&](int t) {
    const int buf   = t & 1;
    const int phase = t >> 5;
    const int kk    = (t & 31) << 5;           // K offset within H
    const float* Ap = phase ? hprev : dt;
    const int rem   = wave * 32 + lane;        // 0..255
    const int arow  = rem >> 3;                // 0..31
    const int akq   = rem & 7;
#pragma unroll
    for (int i = 0; i < 2; ++i) {              // A: 64 rows x 8 chunks
      const int row = arow + i * 32;
      async_copy16(lds_off(&As[buf][row * LDA + (akq << 2)]),
                   Ap + (size_t)(m0 + row) * Hdim + kk + (akq << 2));
    }
#pragma unroll
    for (int g = 0; g < 5; ++g) {              // B: per gate 32 rows x 8 chunks
      const float* Bsrc = phase ? gp.U[g] : gp.W[g];
      async_copy16(lds_off(&Bs[buf][(g * BN + arow) * LDB + (akq << 2)]),
                   Bsrc + (size_t)(n0 + arow) * Hdim + kk + (akq << 2));
    }
  };

  issue(0);
  for (int t = 0; t < NT; ++t) {
    if (t + 1 < NT) { issue(t + 1); wait_async_le7(); }  // tile t landed
    else            { wait_async_le0(); }
    __syncthreads();                                     // all waves' data visible

    const int buf = t & 1;
    // A frag (16x4 f32): lane -> m = l16, VGPR v -> k = lh*2 + v  (ds_load_b64)
    // B frag (4x16 f32): lane -> n = l16, VGPR v -> k = lh*2 + v
    const float* aBase = &As[buf][(wm * 16 + l16) * LDA + lh * 2];
    const float* bBase = &Bs[buf][(wn * 16 + l16) * LDB + lh * 2];
#pragma unroll
    for (int ks = 0; ks < KT; ks += 4) {
      const v2f a = *(const v2f*)(aBase + ks);
#pragma unroll
      for (int g = 0; g < 5; ++g) {
        const v2f b = *(const v2f*)(bBase + g * (BN * LDB) + ks);
        acc[g] = __builtin_amdgcn_wmma_f32_16x16x4_f32(
            false, a, false, b, (short)0, acc[g], false, false);
      }
    }
    __syncthreads();   // protect buffer before tile t+2 overwrites it
  }

  // --- Fused epilogue: activations + c_t; stash g_t, o_t for finalize ---
  // C/D layout: lane -> n = l16 ; VGPR v -> m = v + 8*lh
  const size_t mBase = (size_t)(m0 + wm * 16 + 8 * lh);
#pragma unroll
  for (int v = 0; v < 8; ++v) {
    const size_t off = (mBase + v) * Hdim + nG;
    const float it = fast_sigmoid(acc[0][v]);   // i
    const float ft = fast_sigmoid(acc[1][v]);   // f
    const float ot = fast_sigmoid(acc[2][v]);   // o
    const float ch = fast_tanh   (acc[3][v]);   // c_hat
    const float gt = fast_sigmoid(acc[4][v]);   // g
    const float cp = cprev[off];
    const float ct = it * ch + ft * cp;
    out_c[off] = ct;
    ws_g[off]  = gt;
    ws_o[off]  = ot;
  }
}

// ---------------------------------------------------------------------------
// Kernel 2: per-row reduction + final blend.
//   One block per row: num = sum(c_t*c_prev), den = sum(c_prev^2),
//   h_t = o * tanh(c_t - g * (num/den) * c_prev).
// ---------------------------------------------------------------------------
__global__ __launch_bounds__(256)
void ortho_lstm_finalize(const float* __restrict__ cprev,
                         const float* __restrict__ ws_g,
                         const float* __restrict__ ws_o,
                         float* __restrict__ out)   // [0..BH) = h_t, [BH..2BH) = c_t
{
  const int m   = blockIdx.x;
  const int tid = threadIdx.x;
  const float* ct_row = out + (size_t)Bdim * Hdim + (size_t)m * Hdim;
  const float* cp_row = cprev + (size_t)m * Hdim;

  float ctv[4], cpv[4];
  float num = 0.0f, den = 0.0f;
#pragma unroll
  for (int j = 0; j < 4; ++j) {
    const int n = tid + (j << 8);
    ctv[j] = ct_row[n];
    cpv[j] = cp_row[n];
    num += ctv[j] * cpv[j];
    den += cpv[j] * cpv[j];
  }

  __shared__ float sn[256];
  __shared__ float sd[256];
  sn[tid] = num; sd[tid] = den;
  __syncthreads();
  for (int s = 128; s > 0; s >>= 1) {
    if (tid < s) { sn[tid] += sn[tid + s]; sd[tid] += sd[tid + s]; }
    __syncthreads();
  }
  const float frac = sn[0] / sd[0];

#pragma unroll
  for (int j = 0; j < 4; ++j) {
    const int n = tid + (j << 8);
    const size_t off = (size_t)m * Hdim + n;
    const float cdiv = ctv[j] - ws_g[off] * frac * cpv[j];
    out[off] = ws_o[off] * fast_tanh(cdiv);
  }
}

// ---------------------------------------------------------------------------
extern "C" void kernel_launch(void* const* d_in, const int* in_sizes, int n_in,
                              void* d_out, int out_size, void* d_ws, size_t ws_size,
                              hipStream_t stream) {
  (void)in_sizes; (void)n_in; (void)out_size; (void)ws_size;

  const float* dt = (const float*)d_in[0];
  const float* hp = (const float*)d_in[1];
  const float* cp = (const float*)d_in[2];

  GatePtrs gp;
  for (int g = 0; g < 5; ++g) {
    gp.W[g] = (const float*)d_in[3 + 3 * g];
    gp.b[g] = (const float*)d_in[4 + 3 * g];
    gp.U[g] = (const float*)d_in[5 + 3 * g];
  }

  float* out   = (float*)d_out;
  float* out_c = out + (size_t)Bdim * Hdim;          // second output half = c_t
  float* ws_g  = (float*)d_ws;                       // g_t   (64 MB)
  float* ws_o  = ws_g + (size_t)Bdim * Hdim;         // o_t   (64 MB)

  dim3 grid(Hdim / BN, Bdim / BM);                   // 32 x 256 = 8192 blocks
  ortho_lstm_gates_gemm<<<grid, 256, 0, stream>>>(dt, hp, cp, gp, out_c, ws_g, ws_o);
  ortho_lstm_finalize<<<Bdim, 256, 0, stream>>>(cp, ws_g, ws_o, out);
}